// PolyclassFTTlayer_54004918780530
// MI455X (gfx1250) — compile-verified
//
#include <hip/hip_runtime.h>

// MI455X / gfx1250: wave32, WMMA (V_WMMA_F32_16X16X4_F32 for f32 matmul).
//
// Reference:
//   V[k]   = einsum('s,kasb->kab', z, tt_mid)   (6, 64, 64)   <- 1.61 GB stream, BW-bound
//   vlast  = einsum('s,asb->ab',  z, tt_last)   (64, 1)       <- 4 MB
//   f = tt_first[0]; f = f @ V[0] @ ... @ V[5] @ vlast        <- per-row independent chain
//
// Plan:
//   k_vmid_partial : stream tt_mid with NT float4 loads, per-chunk partial sums (deterministic)
//   k_vlast        : 64 dot-products z . tt_last[a,:]
//   k_reduce       : fixed-order reduction of chunk partials -> V
//   k_chain        : per-16-row-tile WMMA chain (f32 16x16x4) + final dot with vlast

typedef float v2f __attribute__((ext_vector_type(2)));
typedef float v4f __attribute__((ext_vector_type(4)));
typedef float v8f __attribute__((ext_vector_type(8)));

#define S_DIM 16384
#define RANK  64
#define NMID  6
#define NC    16               // s-chunks for the streaming contraction
#define CH    (S_DIM / NC)     // 1024

// workspace layout (in floats)
#define V_SZ      (NMID * RANK * RANK)     // 24576
#define PART_OFF  0
#define PART_SZ   (NC * V_SZ)              // 393216 (1.5 MB)
#define V_OFF     PART_SZ
#define VLAST_OFF (V_OFF + V_SZ)

// ---------------------------------------------------------------------------
// Phase 1: partial V[k][a][b] per s-chunk.  grid = (NC, 64, 6), block = 256.
// Block layout: 16 (tx: b-quad) x 16 (ty: s-subrow). Each thread streams a
// float4 column strip with non-temporal loads (data is touch-once, > L2).
// ---------------------------------------------------------------------------
__global__ __launch_bounds__(256) void k_vmid_partial(const float* __restrict__ z,
                                                      const float* __restrict__ ttm,
                                                      float* __restrict__ part) {
  const int tx = threadIdx.x & 15;   // b = 4*tx .. 4*tx+3
  const int ty = threadIdx.x >> 4;   // s sub-row
  const int chunk = blockIdx.x, a = blockIdx.y, k = blockIdx.z;
  const int s0 = chunk * CH;

  const float* base = ttm + (((size_t)(k * RANK + a)) * S_DIM + s0) * RANK + tx * 4;
  const float* zb = z + s0;

  v4f acc = {0.f, 0.f, 0.f, 0.f};
  for (int s = ty; s < CH; s += 16) {
    float zs = zb[s];
    v4f v = __builtin_nontemporal_load((const v4f*)(base + (size_t)s * RANK));
    acc += v * zs;
  }

  __shared__ v4f red[256];
  red[threadIdx.x] = acc;
  __syncthreads();
  for (int str = 128; str >= 16; str >>= 1) {
    if (threadIdx.x < str) red[threadIdx.x] += red[threadIdx.x + str];
    __syncthreads();
  }
  if (threadIdx.x < 16) {
    *(v4f*)&part[(size_t)chunk * V_SZ + (size_t)(k * RANK + a) * RANK + threadIdx.x * 4] =
        red[threadIdx.x];
  }
}

// ---------------------------------------------------------------------------
// Phase 1b: vlast[a] = z . tt_last[a, :, 0].  grid = 64, block = 256.
// ---------------------------------------------------------------------------
__global__ __launch_bounds__(256) void k_vlast(const float* __restrict__ z,
                                               const float* __restrict__ ttl,
                                               float* __restrict__ vlast) {
  const int a = blockIdx.x;
  const v4f* z4 = (const v4f*)z;
  const v4f* b4 = (const v4f*)(ttl + (size_t)a * S_DIM);
  v4f acc4 = {0.f, 0.f, 0.f, 0.f};
  for (int i = threadIdx.x; i < S_DIM / 4; i += 256)
    acc4 += z4[i] * __builtin_nontemporal_load(&b4[i]);
  float acc = acc4.x + acc4.y + acc4.z + acc4.w;

  __shared__ float red[256];
  red[threadIdx.x] = acc;
  __syncthreads();
  for (int str = 128; str >= 1; str >>= 1) {
    if (threadIdx.x < str) red[threadIdx.x] += red[threadIdx.x + str];
    __syncthreads();
  }
  if (threadIdx.x == 0) vlast[a] = red[0];
}

// ---------------------------------------------------------------------------
// Phase 2: fixed-order chunk reduction (deterministic, no fp atomics).
// ---------------------------------------------------------------------------
__global__ __launch_bounds__(256) void k_reduce(const float* __restrict__ part,
                                                float* __restrict__ V) {
  const int i = blockIdx.x * 256 + threadIdx.x;   // 0 .. V_SZ-1
  float s = 0.f;
#pragma unroll
  for (int c = 0; c < NC; ++c) s += part[(size_t)c * V_SZ + i];
  V[i] = s;
}

// ---------------------------------------------------------------------------
// Phase 3: WMMA chain.  grid = 128 blocks x 256 threads (8 waves).
// Each wave owns a 16x64 tile of f and runs all 6 chain steps locally using
// V_WMMA_F32_16X16X4_F32 (native fp32 matrix op, preserves f32 semantics).
//
// Fragment layouts (ISA 7.12.2, wave32):
//   A 16x4 f32 : lane m=lane&15, kh=lane>>4; v0=F[m][k0+2kh], v1=F[m][k0+2kh+1]
//   B 4x16 f32 : lane n=lane&15, kh=lane>>4; v0=V[k0+2kh][n], v1=V[k0+2kh+1][n]
//   D 16x16    : vgpr r: D[r + 8*(lane>>4)][lane&15]
// D->A conversion goes through per-wave LDS scratch (DS ops in-order per wave).
// ---------------------------------------------------------------------------
__global__ __launch_bounds__(256) void k_chain(const float* __restrict__ f0,
                                               const float* __restrict__ V,
                                               const float* __restrict__ vlast,
                                               float* __restrict__ out) {
  __shared__ float sV[RANK * RANK];      // 16 KB: current V[k]
  __shared__ float sVlast[RANK];
  __shared__ float sF[8][16 * RANK];     // 32 KB: per-wave 16x64 f scratch

  const int tid  = threadIdx.x;
  const int wave = tid >> 5;
  const int lane = tid & 31;
  const int m  = lane & 15;
  const int kh = lane >> 4;
  float* scr = &sF[wave][0];

  const int row0 = blockIdx.x * 128 + wave * 16;

  if (tid < RANK) sVlast[tid] = vlast[tid];

  // coalesced load of this wave's 16x64 tile of tt_first into LDS scratch
  {
    const v4f* src = (const v4f*)(f0 + (size_t)row0 * RANK);
    for (int i = lane; i < (16 * RANK) / 4; i += 32) ((v4f*)scr)[i] = src[i];
  }

  // A fragments from scratch
  v2f afrag[16];
#pragma unroll
  for (int c = 0; c < 16; ++c)
    afrag[c] = *(const v2f*)&scr[m * RANK + 4 * c + 2 * kh];

  for (int k = 0; k < NMID; ++k) {
    __syncthreads();  // previous sV consumers done (also makes sVlast/tile visible)
    {
      const v4f* src = (const v4f*)(V + (size_t)k * RANK * RANK);
      for (int i = tid; i < (RANK * RANK) / 4; i += 256) ((v4f*)sV)[i] = src[i];
    }
    __syncthreads();

#pragma unroll
    for (int j = 0; j < 4; ++j) {        // 16-column output blocks
      v8f acc = {0.f, 0.f, 0.f, 0.f, 0.f, 0.f, 0.f, 0.f};
#pragma unroll
      for (int c = 0; c < 16; ++c) {     // K = 64 in chunks of 4
        v2f b;
        b.x = sV[(4 * c + 2 * kh) * RANK + 16 * j + m];
        b.y = sV[(4 * c + 2 * kh + 1) * RANK + 16 * j + m];
        acc = __builtin_amdgcn_wmma_f32_16x16x4_f32(false, afrag[c], false, b,
                                                    (short)0, acc, false, false);
      }
#pragma unroll
      for (int r = 0; r < 8; ++r)
        scr[(r + 8 * kh) * RANK + 16 * j + m] = acc[r];
    }

    // intra-wave LDS store->load is in-order on CDNA5; just stop compiler reordering
    asm volatile("" ::: "memory");
#pragma unroll
    for (int c = 0; c < 16; ++c)
      afrag[c] = *(const v2f*)&scr[m * RANK + 4 * c + 2 * kh];
  }

  // final: out[row0+m] = sum_j F[m][j] * vlast[j]
  if (lane < 16) {
    float s = 0.f;
#pragma unroll 8
    for (int j = 0; j < RANK; ++j) s += scr[m * RANK + j] * sVlast[j];
    out[row0 + m] = s;
  }
}

// ---------------------------------------------------------------------------
extern "C" void kernel_launch(void* const* d_in, const int* in_sizes, int n_in,
                              void* d_out, int out_size, void* d_ws, size_t ws_size,
                              hipStream_t stream) {
  (void)in_sizes; (void)n_in; (void)out_size; (void)ws_size;
  const float* z  = (const float*)d_in[0];
  const float* tf = (const float*)d_in[1];   // (1, S, 64)
  const float* tm = (const float*)d_in[2];   // (6, 64, S, 64)
  const float* tl = (const float*)d_in[3];   // (64, S, 1)
  float* ws    = (float*)d_ws;               // needs ~1.67 MB
  float* part  = ws + PART_OFF;
  float* V     = ws + V_OFF;
  float* vlast = ws + VLAST_OFF;
  float* out   = (float*)d_out;

  k_vmid_partial<<<dim3(NC, RANK, NMID), 256, 0, stream>>>(z, tm, part);
  k_vlast<<<dim3(RANK), 256, 0, stream>>>(z, tl, vlast);
  k_reduce<<<dim3(V_SZ / 256), 256, 0, stream>>>(part, V);
  k_chain<<<dim3(S_DIM / 128), 256, 0, stream>>>(tf, V, vlast, out);
}